// POM1bLoss_72430328479936
// MI455X (gfx1250) — compile-verified
//
#include <hip/hip_runtime.h>
#include <hip/hip_bf16.h>

#define NCLS 7
#define BLOCK 256
#define TILE_ROWS 256
#define TILE_DW (TILE_ROWS * NCLS)   /* 1792 dwords = 7168 B per tile */
#define NBLOCKS 1024
#define EPSF 1e-10f

// ---------------- CDNA5 async global->LDS helpers ----------------

typedef __attribute__((address_space(1))) int as1_int;   // global
typedef __attribute__((address_space(3))) int as3_int;   // LDS

__device__ __forceinline__ void async_copy_dword(const float* g, float* l) {
#if __has_builtin(__builtin_amdgcn_global_load_async_to_lds_b32)
  __builtin_amdgcn_global_load_async_to_lds_b32(
      (as1_int*)g, (as3_int*)l, /*offset=*/0, /*cpol=*/0);
#else
  unsigned lds_off = (unsigned)(__UINTPTR_TYPE__)(as3_int*)l;
  asm volatile("global_load_async_to_lds_b32 %0, %1, off"
               :: "v"(lds_off), "v"(g) : "memory");
#endif
}

template <int N>
__device__ __forceinline__ void wait_asynccnt() {
#if __has_builtin(__builtin_amdgcn_s_wait_asynccnt)
  __builtin_amdgcn_s_wait_asynccnt(N);
#else
  asm volatile("s_wait_asynccnt %0" :: "n"(N) : "memory");
#endif
}

// ---------------- per-row loss ----------------

__device__ __forceinline__ float row_loss(const float* __restrict__ x, int t) {
  float m = x[0];
#pragma unroll
  for (int k = 1; k < NCLS; ++k) m = fmaxf(m, x[k]);
  float denom = 0.f, num = 0.f;
#pragma unroll
  for (int k = 0; k < NCLS; ++k) {
    float ev = __expf(x[k] - m);          // v_exp_f32
    denom += ev;
    int dd = k - t;                        // band |k - t| <= 1, branch-free
    num += (dd >= -1 && dd <= 1) ? ev : 0.f;
  }
  return -__logf(num / denom + EPSF);      // v_log_f32
}

// ---------------- main streaming kernel ----------------

__global__ void __launch_bounds__(BLOCK)
pom1b_main(const float* __restrict__ logits, const int* __restrict__ targets,
           float* __restrict__ partials, int nrows) {
  __shared__ float tile[2][TILE_DW];
  __shared__ float red[BLOCK];

  const int tid = threadIdx.x;
  const int nTiles = nrows / TILE_ROWS;
  float acc = 0.f;

  int it = blockIdx.x;
  int buf = 0;

  // prologue: stage first tile into buffer 0
  if (it < nTiles) {
    const float* gbase = logits + (size_t)it * TILE_DW;
#pragma unroll
    for (int k = 0; k < NCLS; ++k) {
      const int d = tid + k * BLOCK;
      async_copy_dword(gbase + d, &tile[0][d]);
    }
  }

  for (; it < nTiles; it += gridDim.x) {
    const int nxt = it + gridDim.x;
    const bool hasNext = nxt < nTiles;
    if (hasNext) {
      const float* gbase = logits + (size_t)nxt * TILE_DW;
#pragma unroll
      for (int k = 0; k < NCLS; ++k) {
        const int d = tid + k * BLOCK;
        async_copy_dword(gbase + d, &tile[buf ^ 1][d]);
      }
      wait_asynccnt<NCLS>();   // current tile's 7 loads complete (in-order)
    } else {
      wait_asynccnt<0>();
    }
    __syncthreads();           // cross-wave: everyone's async data landed

    const int row = it * TILE_ROWS + tid;
    acc += row_loss(&tile[buf][tid * NCLS], targets[row]);

    __syncthreads();           // all reads done before this buffer is refilled
    buf ^= 1;
  }

  // remainder rows (nrows % 256), direct global path
  for (int row = nTiles * TILE_ROWS + blockIdx.x * BLOCK + tid; row < nrows;
       row += gridDim.x * BLOCK) {
    float xr[NCLS];
#pragma unroll
    for (int k = 0; k < NCLS; ++k) xr[k] = logits[(size_t)row * NCLS + k];
    acc += row_loss(xr, targets[row]);
  }

  // deterministic block reduction
  red[tid] = acc;
  __syncthreads();
#pragma unroll
  for (int s = BLOCK / 2; s > 0; s >>= 1) {
    if (tid < s) red[tid] += red[tid + s];
    __syncthreads();
  }
  if (tid == 0) partials[blockIdx.x] = red[0];
}

// ---------------- final reduction ----------------

__global__ void __launch_bounds__(BLOCK)
pom1b_reduce(const float* __restrict__ partials, float* __restrict__ out,
             int n, float invB) {
  __shared__ float red[BLOCK];
  float a = 0.f;
  for (int i = threadIdx.x; i < n; i += BLOCK) a += partials[i];
  red[threadIdx.x] = a;
  __syncthreads();
#pragma unroll
  for (int s = BLOCK / 2; s > 0; s >>= 1) {
    if (threadIdx.x < s) red[threadIdx.x] += red[threadIdx.x + s];
    __syncthreads();
  }
  if (threadIdx.x == 0) out[0] = red[0] * invB;
}

// ---------------- launcher ----------------

extern "C" void kernel_launch(void* const* d_in, const int* in_sizes, int n_in,
                              void* d_out, int out_size, void* d_ws, size_t ws_size,
                              hipStream_t stream) {
  const float* logits  = (const float*)d_in[0];
  const int*   targets = (const int*)d_in[1];
  const int    nrows   = in_sizes[1];           // BATCH (targets count)

  int nblocks = NBLOCKS;
  if (ws_size < (size_t)nblocks * sizeof(float))
    nblocks = (int)(ws_size / sizeof(float));
  if (nblocks < 1) nblocks = 1;

  float* partials = (float*)d_ws;

  pom1b_main<<<nblocks, BLOCK, 0, stream>>>(logits, targets, partials, nrows);
  pom1b_reduce<<<1, BLOCK, 0, stream>>>(partials, (float*)d_out, nblocks,
                                        1.0f / (float)nrows);
}